// LearnableAdjacency_57775900066444
// MI455X (gfx1250) — compile-verified
//
#include <hip/hip_runtime.h>
#include <math.h>

typedef float v2f __attribute__((ext_vector_type(2)));
typedef float v8f __attribute__((ext_vector_type(8)));

#define NPTS 8192
#define DIM  512
#define INV_2SIG2 (1.0f / 5000.0f)   // 1/(2*50^2)

// ---------------------------------------------------------------------------
// Kernel 1: row-normalize features (F.normalize eps=1e-12) into workspace.
// Also writes lam = sigmoid(lambda_weight) into the tuple's 2nd output slot.
// ---------------------------------------------------------------------------
__global__ void norm_rows_kernel(const float* __restrict__ feat,
                                 const float* __restrict__ lambda_w,
                                 float* __restrict__ fn,
                                 float* __restrict__ lam_out) {
    __shared__ float red[256];
    const int row = blockIdx.x;
    const int tid = threadIdx.x;
    const float* f = feat + (size_t)row * DIM;
    float a0 = f[tid];
    float a1 = f[tid + 256];
    red[tid] = a0 * a0 + a1 * a1;
    __syncthreads();
    for (int s = 128; s > 0; s >>= 1) {
        if (tid < s) red[tid] += red[tid + s];
        __syncthreads();
    }
    const float scale = 1.0f / fmaxf(sqrtf(red[0]), 1e-12f);
    float* o = fn + (size_t)row * DIM;
    o[tid]       = a0 * scale;
    o[tid + 256] = a1 * scale;
    if (row == 0 && tid == 0) {
        lam_out[0] = 1.0f / (1.0f + expf(-lambda_w[0]));
    }
}

// ---------------------------------------------------------------------------
// Kernel 2: adj = sigmoid((lam*sim + (1-lam)*spatial) * temperature)
// sim tile via V_WMMA_F32_16X16X4_F32. One 16x16 tile per wave32,
// 8 waves (256 threads) per block.
//
// fp32 WMMA lane layout (ISA 7.12.2):
//   A (16x4):  lanes 0-15 hold M=lane, K={0,1}; lanes 16-31 hold M=lane-16, K={2,3}
//   B (4x16):  lanes 0-15 hold N=lane, K={0,1}; lanes 16-31 hold N=lane-16, K={2,3}
//   C/D (16x16): vgpr v, lane<16 -> (M=v,   N=lane)
//                         lane>=16 -> (M=v+8, N=lane-16)
// Since B = Fn^T tile, B[k][n] = Fn[j0+n][k]: both A and B load as one
// float2 (global_load_b64) per lane per K-step.
// ---------------------------------------------------------------------------
__global__ void adj_wmma_kernel(const float* __restrict__ fn,
                                const float* __restrict__ pos,
                                const float* __restrict__ lambda_w,
                                const float* __restrict__ temp_p,
                                float* __restrict__ out) {
    const int lane = threadIdx.x & 31;
    const int wave = threadIdx.x >> 5;
    const int tile = blockIdx.x * 8 + wave;       // 512*512 tiles total
    const int ti = tile >> 9;                     // tile / 512
    const int tj = tile & 511;                    // tile % 512
    const int i0 = ti * 16;
    const int j0 = tj * 16;

    const int mrow = lane & 15;
    const int koff = (lane >> 4) * 2;             // lanes 16-31 take K+2,K+3
    const float* pa = fn + (size_t)(i0 + mrow) * DIM + koff;
    const float* pb = fn + (size_t)(j0 + mrow) * DIM + koff;

    v8f c = {};
#pragma unroll 4
    for (int k = 0; k < DIM; k += 4) {
        v2f a = *(const v2f*)(pa + k);
        v2f b = *(const v2f*)(pb + k);
        c = __builtin_amdgcn_wmma_f32_16x16x4_f32(
                /*neg_a=*/false, a, /*neg_b=*/false, b,
                /*c_mod=*/(short)0, c, /*reuse_a=*/false, /*reuse_b=*/false);
    }

    const float lam  = 1.0f / (1.0f + expf(-lambda_w[0]));
    const float temp = temp_p[0];

    const int col = j0 + (lane & 15);
    const float cx = pos[2 * col];
    const float cy = pos[2 * col + 1];
    const int rbase = i0 + ((lane >> 4) << 3);    // +8 for upper half-wave

#pragma unroll
    for (int v = 0; v < 8; v++) {
        const int row = rbase + v;
        const float dx = pos[2 * row]     - cx;
        const float dy = pos[2 * row + 1] - cy;
        const float spatial = expf(-(dx * dx + dy * dy) * INV_2SIG2);
        float val = (lam * c[v] + (1.0f - lam) * spatial) * temp;
        out[(size_t)row * NPTS + col] = 1.0f / (1.0f + expf(-val));
    }
}

// ---------------------------------------------------------------------------
// Kernel 3: per-row top-k mask. One block (256 threads) per row.
// Row (32KB) lives in LDS. k iterations of: masked block max-reduce, then
// lowest-index argmax (matches top_k tie-breaking), then set mask bit.
// Finally: out = bit ? adj : 0.
// ---------------------------------------------------------------------------
__global__ void topk_mask_kernel(float* __restrict__ out,
                                 const int* __restrict__ kptr) {
    __shared__ float    vals[NPTS];
    __shared__ unsigned mask[NPTS / 32];
    __shared__ float    red[256];
    __shared__ int      selIdx;

    const int row = blockIdx.x;
    const int tid = threadIdx.x;
    float* rowp = out + (size_t)row * NPTS;

#pragma unroll
    for (int i = 0; i < 32; i++) vals[i * 256 + tid] = rowp[i * 256 + tid];
    mask[tid] = 0u;
    __syncthreads();

    const int k = kptr[0];
    for (int t = 0; t < k; t++) {
        // masked local max over this thread's 32 elements
        float m = -INFINITY;
#pragma unroll
        for (int i = 0; i < 32; i++) {
            const int idx = i * 256 + tid;
            if (!((mask[idx >> 5] >> (idx & 31)) & 1u)) m = fmaxf(m, vals[idx]);
        }
        red[tid] = m;
        __syncthreads();
        for (int s = 128; s > 0; s >>= 1) {
            if (tid < s) red[tid] = fmaxf(red[tid], red[tid + s]);
            __syncthreads();
        }
        const float maxv = red[0];
        if (tid == 0) selIdx = 0x7FFFFFFF;
        __syncthreads();

        // lowest unmasked index holding maxv (bitwise-exact: maxv is one of vals)
        int best = 0x7FFFFFFF;
        for (int i = 0; i < 32; i++) {
            const int idx = i * 256 + tid;
            if (vals[idx] == maxv && !((mask[idx >> 5] >> (idx & 31)) & 1u)) {
                best = idx;
                break;  // idx increases with i, first hit is this thread's min
            }
        }
        if (best != 0x7FFFFFFF) atomicMin(&selIdx, best);
        __syncthreads();
        if (tid == 0) {
            const int si = selIdx;
            mask[si >> 5] |= (1u << (si & 31));
        }
        __syncthreads();
    }

#pragma unroll
    for (int i = 0; i < 32; i++) {
        const int idx = i * 256 + tid;
        const unsigned bit = (mask[idx >> 5] >> (idx & 31)) & 1u;
        rowp[idx] = bit ? vals[idx] : 0.0f;
    }
}

// ---------------------------------------------------------------------------
extern "C" void kernel_launch(void* const* d_in, const int* in_sizes, int n_in,
                              void* d_out, int out_size, void* d_ws, size_t ws_size,
                              hipStream_t stream) {
    (void)in_sizes; (void)n_in; (void)out_size; (void)ws_size;
    const float* feat  = (const float*)d_in[0];   // (8192, 512) f32
    const float* pos   = (const float*)d_in[1];   // (8192, 2)   f32
    const float* lamw  = (const float*)d_in[2];   // scalar f32
    const float* tempp = (const float*)d_in[3];   // scalar f32
    const int*   kptr  = (const int*)d_in[4];     // scalar int (k=20)

    float* out = (float*)d_out;                   // 8192*8192 adj*mask, then lam
    float* fn  = (float*)d_ws;                    // 16 MB normalized features

    norm_rows_kernel<<<NPTS, 256, 0, stream>>>(feat, lamw, fn,
                                               out + (size_t)NPTS * NPTS);

    const int ntiles = (NPTS / 16) * (NPTS / 16); // 262144
    adj_wmma_kernel<<<ntiles / 8, 256, 0, stream>>>(fn, pos, lamw, tempp, out);

    topk_mask_kernel<<<NPTS, 256, 0, stream>>>(out, kptr);
}